// UnlikelihoodLoss_28054726377865
// MI455X (gfx1250) — compile-verified
//
#include <hip/hip_runtime.h>
#include <hip/hip_bf16.h>
#include <math.h>

// ---------------------------------------------------------------------------
// UnlikelihoodLoss: per-row logsumexp over 50257 fp32 logits ->
// log(1 - p_tgt + eps) -> masked mean. Pure HBM streaming: 823 MB once,
// ~35 us @ 23.3 TB/s. Strategy: CDNA5 async global->LDS copies, 4-deep
// pipeline (3 outstanding DMAs), fully static buffer rotation via unroll-4,
// online logsumexp (1 v_exp_f32 per element).
// ---------------------------------------------------------------------------

#define N_ROWS      4096
#define VOCAB_SZ    50257
#define BLOCK_T     256
#define CHUNK_F     (BLOCK_T * 4)          // 1024 floats = 4 KB per chunk
#define NFULL       (VOCAB_SZ / CHUNK_F)   // 49 full chunks = 50176 floats
#define TAIL_BASE   (NFULL * CHUNK_F)      // remaining 81 floats
#define GROUPS      ((NFULL - 5) / 4)      // 11 unrolled groups of 4 chunks
#define IGN_IDX     (-100)
#define EPS_F       1e-10f

static_assert(NFULL == 49, "pipeline schedule assumes 49 full chunks");
static_assert(GROUPS * 4 + 5 == NFULL, "prologue(3)+loop(44)+epilogue(5)");

// -------- CDNA5 async global->LDS copy (gfx1250), per-lane 16B, GV mode ----
// vdst = per-lane LDS byte address, vaddr = 64-bit global address pair,
// byte displacement folded into the signed-24-bit offset immediate.
template <int OFF>
__device__ __forceinline__ void async_load_b128(unsigned lds_byte_addr,
                                                const void* gaddr) {
  asm volatile("global_load_async_to_lds_b128 %0, %1, off offset:%2"
               :: "v"(lds_byte_addr), "v"(gaddr), "i"(OFF)
               : "memory");
}

template <int CNT>
__device__ __forceinline__ void wait_asynccnt() {
  asm volatile("s_wait_asynccnt %0" :: "i"(CNT) : "memory");
}

// -------- online logsumexp combine: (m,s) x (om,os) -> (m,s) ---------------
__device__ __forceinline__ void lse_combine(float& m, float& s,
                                            float om, float os) {
  float M = fmaxf(m, om);
  s = s * __expf(m - M) + os * __expf(om - M);
  m = M;
}

__global__ __launch_bounds__(BLOCK_T)
void row_lse_kernel(const float* __restrict__ pred,
                    const int* __restrict__ tgt,
                    float* __restrict__ per_row,
                    float* __restrict__ valid_f) {
  __shared__ float4 buf[4][BLOCK_T];          // 16 KB quad buffer
  __shared__ float red_m[BLOCK_T / 32];
  __shared__ float red_s[BLOCK_T / 32];

  const int tid = threadIdx.x;
  const int row = blockIdx.x;
  const float* rp = pred + (size_t)row * VOCAB_SZ;

  // Per-lane LDS byte addresses (flat LDS pointer truncates to the 32-bit
  // DS address per ISA 10.2). Buffers are 4096 B apart -> folds to adds.
  const unsigned l0 = (unsigned)(size_t)(&buf[0][tid]);
  const unsigned l1 = (unsigned)(size_t)(&buf[1][tid]);
  const unsigned l2 = (unsigned)(size_t)(&buf[2][tid]);
  const unsigned l3 = (unsigned)(size_t)(&buf[3][tid]);

  float m = -__builtin_inff();
  float s = 0.0f;

  // One pipelined chunk consume: rescale to new max, pairwise exp sum.
  auto consume = [&](const float4& x) {
    const float m4 = fmaxf(fmaxf(x.x, x.y), fmaxf(x.z, x.w));
    if (m4 > m) { s *= __expf(m - m4); m = m4; }
    const float e0 = __expf(x.x - m);
    const float e1 = __expf(x.y - m);
    const float e2 = __expf(x.z - m);
    const float e3 = __expf(x.w - m);
    s += (e0 + e1) + (e2 + e3);
  };

  // ---- prologue: prime 3 chunks (chunks 0,1,2 -> buf0,1,2) ----------------
  const float* gp = rp + tid * 4;             // this lane's chunk-0 slot
  async_load_b128<0>(l0, gp);
  async_load_b128<4096>(l1, gp);
  async_load_b128<8192>(l2, gp);
  const float* cur = gp + 3 * CHUNK_F;        // lane slot of chunk 3

  // ---- steady state: 44 chunks, 3 DMAs always in flight -------------------
  for (int g = 0; g < GROUPS; ++g) {
    async_load_b128<0>(l3, cur);              // chunk 4g+3 -> buf3
    wait_asynccnt<3>();
    consume(buf[0][tid]);                     // chunk 4g+0

    async_load_b128<4096>(l0, cur);           // chunk 4g+4 -> buf0
    wait_asynccnt<3>();
    consume(buf[1][tid]);                     // chunk 4g+1

    async_load_b128<8192>(l1, cur);           // chunk 4g+5 -> buf1
    wait_asynccnt<3>();
    consume(buf[2][tid]);                     // chunk 4g+2

    async_load_b128<12288>(l2, cur);          // chunk 4g+6 -> buf2
    wait_asynccnt<3>();
    consume(buf[3][tid]);                     // chunk 4g+3

    cur += 4 * CHUNK_F;                       // advance 16 KB
  }

  // ---- epilogue: chunks 44..48 (cur points at lane slot of chunk 47) ------
  async_load_b128<0>(l3, cur);                // chunk 47 -> buf3
  wait_asynccnt<3>();
  consume(buf[0][tid]);                       // chunk 44
  async_load_b128<4096>(l0, cur);             // chunk 48 -> buf0
  wait_asynccnt<3>();
  consume(buf[1][tid]);                       // chunk 45
  wait_asynccnt<2>();
  consume(buf[2][tid]);                       // chunk 46
  wait_asynccnt<1>();
  consume(buf[3][tid]);                       // chunk 47
  wait_asynccnt<0>();
  consume(buf[0][tid]);                       // chunk 48

  // ---- tail: 81 floats via plain global loads -----------------------------
  for (int i = TAIL_BASE + tid; i < VOCAB_SZ; i += BLOCK_T) {
    const float x = rp[i];
    if (x > m) { s *= __expf(m - x); m = x; }
    s += __expf(x - m);
  }

  // ---- wave32 reduction of (m, s) -----------------------------------------
  #pragma unroll
  for (int off = 16; off > 0; off >>= 1) {
    const float om = __shfl_xor(m, off, 32);
    const float os = __shfl_xor(s, off, 32);
    lse_combine(m, s, om, os);
  }

  const int wave = tid >> 5;
  if ((tid & 31) == 0) { red_m[wave] = m; red_s[wave] = s; }
  __syncthreads();

  if (tid == 0) {
    float M = red_m[0], S = red_s[0];
    #pragma unroll
    for (int w = 1; w < BLOCK_T / 32; ++w) lse_combine(M, S, red_m[w], red_s[w]);

    const float logZ = M + logf(S);
    const int t = tgt[row];
    const bool valid = (t != IGN_IDX);
    int st = t < 0 ? 0 : t;
    if (st > VOCAB_SZ - 1) st = VOCAB_SZ - 1;
    const float tl = rp[st];
    const float prob = __expf(tl - logZ);
    const float pr = logf(1.0f - prob + EPS_F);
    per_row[row] = valid ? pr : 0.0f;
    valid_f[row] = valid ? 1.0f : 0.0f;
  }
}

// -------- deterministic final reduction over 4096 rows ---------------------
__global__ __launch_bounds__(BLOCK_T)
void finalize_kernel(const float* __restrict__ per_row,
                     const float* __restrict__ valid_f,
                     float* __restrict__ out) {
  __shared__ float sh_t[BLOCK_T];
  __shared__ float sh_v[BLOCK_T];
  const int tid = threadIdx.x;

  float t = 0.0f, v = 0.0f;
  for (int i = tid; i < N_ROWS; i += BLOCK_T) {
    t += per_row[i];
    v += valid_f[i];
  }
  sh_t[tid] = t;
  sh_v[tid] = v;
  __syncthreads();

  #pragma unroll
  for (int off = BLOCK_T / 2; off > 0; off >>= 1) {
    if (tid < off) {
      sh_t[tid] += sh_t[tid + off];
      sh_v[tid] += sh_v[tid + off];
    }
    __syncthreads();
  }
  if (tid == 0) out[0] = -sh_t[0] / sh_v[0];
}

extern "C" void kernel_launch(void* const* d_in, const int* in_sizes, int n_in,
                              void* d_out, int out_size, void* d_ws, size_t ws_size,
                              hipStream_t stream) {
  (void)in_sizes; (void)n_in; (void)out_size; (void)ws_size;

  const float* pred = (const float*)d_in[0];
  const int*   tgt  = (const int*)d_in[1];

  float* ws       = (float*)d_ws;
  float* per_row  = ws;             // [N_ROWS]
  float* valid_f  = ws + N_ROWS;    // [N_ROWS]

  row_lse_kernel<<<N_ROWS, BLOCK_T, 0, stream>>>(pred, tgt, per_row, valid_f);
  finalize_kernel<<<1, BLOCK_T, 0, stream>>>(per_row, valid_f, (float*)d_out);
}